// TransducerLoss_53506702573642
// MI455X (gfx1250) — compile-verified
//
#include <hip/hip_runtime.h>
#include <math.h>

#define BB   8
#define MAXT 256
#define MAXU 64
#define AA   512

typedef __attribute__((ext_vector_type(2))) float v2f;
typedef __attribute__((ext_vector_type(8))) float v8f;

// ---------------------------------------------------------------------------
// Phase 1: gather blank & emit log-probs out of the 268MB tensor (1MB useful)
// ---------------------------------------------------------------------------
__global__ void __launch_bounds__(256)
rnnt_gather_kernel(const float* __restrict__ lp, const int* __restrict__ labels,
                   float* __restrict__ blank_ws, float* __restrict__ emit_ws) {
    int idx = blockIdx.x * blockDim.x + threadIdx.x;      // (b*MAXT + t)*MAXU + u
    if (idx >= BB * MAXT * MAXU) return;
    int u  = idx & (MAXU - 1);
    int bt = idx >> 6;
    int b  = bt >> 8;                                     // MAXT == 256
    long base = (long)idx * AA;
    blank_ws[idx] = lp[base];                             // blank == index 0
    float e = 0.0f;
    if (u < MAXU - 1) {
        int lab = labels[b * (MAXU - 1) + u];             // labels in [1, AA)
        e = lp[base + lab];
    }
    emit_ws[idx] = e;
}

// ---------------------------------------------------------------------------
// Phase 2: E[b,t,u] = sum_{k<u} emit[b,t,k]  ==  Lexcl(64x64) @ emit(64x256)
// per batch, via V_WMMA_F32_16X16X4_F32. One wave == one 16x16 output tile.
// ---------------------------------------------------------------------------
__global__ void __launch_bounds__(32)
rnnt_cumsum_wmma_kernel(const float* __restrict__ emit_ws, float* __restrict__ E_ws) {
    int tile  = blockIdx.x;           // 512 tiles: b(8) x mTile(4) x nTile(16)
    int b     = tile >> 6;
    int mTile = (tile >> 4) & 3;      // u-tile (M)
    int nTile = tile & 15;            // t-tile (N)

    int lane = threadIdx.x;
    int half = lane >> 4;             // 0: lanes 0-15, 1: lanes 16-31
    int l15  = lane & 15;

    int m    = mTile * 16 + l15;      // global u row (A-matrix rows)
    int tcol = nTile * 16 + l15;      // global t col (B/C/D cols)

    const float* ebase = emit_ws + ((long)b * MAXT + tcol) * MAXU;

    v8f acc = {};
    for (int k = 0; k < MAXU; k += 4) {
        int k0 = k + half * 2;        // ISA 32-bit A layout: lanes16-31 carry K+2,K+3
        v2f a, bm;
        a.x  = (k0     < m) ? 1.0f : 0.0f;   // strict lower triangular -> exclusive scan
        a.y  = (k0 + 1 < m) ? 1.0f : 0.0f;
        bm.x = ebase[k0];                     // B[k,n] = emit[b][t=tcol][u=k]
        bm.y = ebase[k0 + 1];
        acc = __builtin_amdgcn_wmma_f32_16x16x4_f32(
            /*neg_a=*/false, a, /*neg_b=*/false, bm,
            /*c_mod=*/(short)0, acc, /*reuse_a=*/false, /*reuse_b=*/false);
    }

    // D layout: VGPR r -> (u = mTile*16 + r + half*8, t = tcol); 8 consecutive u
    float* out = E_ws + ((long)b * MAXT + tcol) * MAXU + mTile * 16 + half * 8;
#pragma unroll
    for (int r = 0; r < 8; ++r) out[r] = acc[r];
}

// ---------------------------------------------------------------------------
// Phase 3: sequential DP over t; logcumsumexp over u via wave32 shuffle scan
// ---------------------------------------------------------------------------
__device__ __forceinline__ float logaddexpf_(float x, float y) {
    float m = fmaxf(x, y);
    float d = fabsf(x - y);
    float r = m + log1pf(__expf(-d));
    return (m == -INFINITY) ? -INFINITY : r;
}

__global__ void __launch_bounds__(64)
rnnt_dp_kernel(const float* __restrict__ blank_ws, const float* __restrict__ E_ws,
               const int* __restrict__ T, const int* __restrict__ U,
               float* __restrict__ logp_ws) {
    int b    = blockIdx.x;
    int u    = threadIdx.x;           // 0..63
    int lane = u & 31;
    int wave = u >> 5;

    __shared__ float s_carry;
    __shared__ float s_alpha[MAXU];

    const float* Eb = E_ws     + (long)b * MAXT * MAXU;
    const float* Bl = blank_ws + (long)b * MAXT * MAXU;
    int Tb = T[b];

    float alpha = Eb[u];              // alpha0 = E[:,0,:]

    for (int t = 1; t < Tb; ++t) {
        float blank_prev = Bl[(t - 1) * MAXU + u];
        float Et         = Eb[t * MAXU + u];
        // prefetch next step's rows into WGP$/L2 path (global_prefetch_b8)
        __builtin_prefetch(&Bl[t * MAXU + u], 0, 3);
        __builtin_prefetch(&Eb[(t + 1) * MAXU + u], 0, 3);

        float x = alpha + blank_prev - Et;

        // inclusive logaddexp scan within wave32
#pragma unroll
        for (int d = 1; d < 32; d <<= 1) {
            float n = __shfl_up(x, d, 32);
            if (lane >= d) x = logaddexpf_(x, n);
        }
        if (wave == 0 && lane == 31) s_carry = x;
        __syncthreads();
        if (wave == 1) x = logaddexpf_(x, s_carry);
        alpha = Et + x;
        __syncthreads();              // protect s_carry before next overwrite
    }

    s_alpha[u] = alpha;
    __syncthreads();
    if (u == 0) {
        int Ub = U[b];                                // in [32, 63]
        float alpha_final = s_alpha[Ub];
        float lp_final    = Bl[(Tb - 1) * MAXU + Ub]; // blank at (T-1, U)
        logp_ws[b] = (alpha_final + lp_final) / (float)Tb;
    }
}

// ---------------------------------------------------------------------------
// Phase 4: loss = -mean(log_p)
// ---------------------------------------------------------------------------
__global__ void __launch_bounds__(32)
rnnt_reduce_kernel(const float* __restrict__ logp_ws, float* __restrict__ out) {
    if (threadIdx.x == 0) {
        float s = 0.0f;
#pragma unroll
        for (int b = 0; b < BB; ++b) s += logp_ws[b];
        out[0] = -s / (float)BB;
    }
}

extern "C" void kernel_launch(void* const* d_in, const int* in_sizes, int n_in,
                              void* d_out, int out_size, void* d_ws, size_t ws_size,
                              hipStream_t stream) {
    const float* lp     = (const float*)d_in[0];  // (B, MAXT, MAXU, A) f32
    const int*   labels = (const int*)  d_in[1];  // (B, MAXU-1)
    const int*   T      = (const int*)  d_in[2];  // (B,)
    const int*   U      = (const int*)  d_in[3];  // (B,)

    const long N = (long)BB * MAXT * MAXU;
    float* blank_ws = (float*)d_ws;
    float* emit_ws  = blank_ws + N;
    float* E_ws     = emit_ws + N;
    float* logp_ws  = E_ws + N;

    rnnt_gather_kernel<<<(int)((N + 255) / 256), 256, 0, stream>>>(lp, labels, blank_ws, emit_ws);
    rnnt_cumsum_wmma_kernel<<<BB * 4 * 16, 32, 0, stream>>>(emit_ws, E_ws);
    rnnt_dp_kernel<<<BB, MAXU, 0, stream>>>(blank_ws, E_ws, T, U, logp_ws);
    rnnt_reduce_kernel<<<1, 32, 0, stream>>>(logp_ws, (float*)d_out);
}